// GraphFeatureAwareGCN_28913719837314
// MI455X (gfx1250) — compile-verified
//
#include <hip/hip_runtime.h>
#include <hip/hip_bf16.h>

#define Nn 50000
#define Ee 800000
#define Ff 128
#define Hh 128
#define Gg 256
#define GF 64

typedef __attribute__((ext_vector_type(2))) float v2f;
typedef __attribute__((ext_vector_type(8))) float v8f;

// ---------------------------------------------------------------------------
// Generic zero-fill
__global__ void fill_zero_kernel(float* __restrict__ p, long n) {
    long i = (long)blockIdx.x * blockDim.x + threadIdx.x;
    if (i < n) p[i] = 0.0f;
}

// ---------------------------------------------------------------------------
// deg[dst[e]] += 1 for each edge
__global__ void edge_deg_kernel(const int* __restrict__ dst, float* __restrict__ deg, int E) {
    int e = blockIdx.x * blockDim.x + threadIdx.x;
    if (e < E) atomicAdd(&deg[dst[e]], 1.0f);
}

// dis[i] = 1/sqrt(deg[i]+1), in place
__global__ void make_dis_kernel(float* __restrict__ deg, int N) {
    int i = blockIdx.x * blockDim.x + threadIdx.x;
    if (i < N) deg[i] = 1.0f / sqrtf(deg[i] + 1.0f);
}

// ---------------------------------------------------------------------------
// C[N,128] = A[N,128] @ W[128,128]  (fp32, V_WMMA_F32_16X16X4_F32)
// One wave per 16-row stripe; 8 column tiles of 16; K loop in steps of 4.
__global__ void gemm_f32_wmma_kernel(const float* __restrict__ A,
                                     const float* __restrict__ W,
                                     float* __restrict__ C, int nRowTiles) {
    int wave = (blockIdx.x * blockDim.x + threadIdx.x) >> 5;
    int lane = threadIdx.x & 31;
    if (wave >= nRowTiles) return;          // wave-uniform: EXEC stays all-1s
    const int m  = lane & 15;               // A row within tile / C col
    const int kh = lane >> 4;               // K half (0 or 1)
    const long row0 = (long)wave * 16;

    v8f acc[8];
#pragma unroll
    for (int j = 0; j < 8; ++j) acc[j] = (v8f){};

    for (int k = 0; k < Ff; k += 4) {
        // A(16x4): lane holds A[row0+m][k + 2*kh + {0,1}]
        const float* ap = A + (row0 + m) * Ff + k + 2 * kh;
        v2f a; a.x = ap[0]; a.y = ap[1];
#pragma unroll
        for (int j = 0; j < 8; ++j) {
            // B(4x16): lane holds W[k + 2*kh + {0,1}][j*16 + m]
            const float* bp = W + (long)(k + 2 * kh) * Hh + j * 16 + m;
            v2f b; b.x = bp[0]; b.y = bp[Hh];
            acc[j] = __builtin_amdgcn_wmma_f32_16x16x4_f32(
                false, a, false, b, (short)0, acc[j], false, false);
        }
    }
    // C/D layout: acc[j][i] = C[row0 + i + 8*kh][j*16 + m]
#pragma unroll
    for (int j = 0; j < 8; ++j) {
#pragma unroll
        for (int i = 0; i < 8; ++i) {
            C[(row0 + i + 8 * kh) * Hh + j * 16 + m] = acc[j][i];
        }
    }
}

// ---------------------------------------------------------------------------
// agg[dst] += dis[src]*dis[dst] * h[src]   (one 32-lane group per edge)
__global__ void scatter_edges_kernel(const float* __restrict__ h,
                                     const float* __restrict__ dis,
                                     const int* __restrict__ src,
                                     const int* __restrict__ dst,
                                     float* __restrict__ agg, int E) {
    long t = (long)blockIdx.x * blockDim.x + threadIdx.x;
    int e = (int)(t >> 5);
    int lane = (int)(t & 31);
    if (e >= E) return;
    int s = src[e], d = dst[e];
    float norm = dis[s] * dis[d];
    float4 v = ((const float4*)(h + (long)s * Hh))[lane];
    float* ap = agg + (long)d * Hh + lane * 4;
    atomicAdd(ap + 0, norm * v.x);
    atomicAdd(ap + 1, norm * v.y);
    atomicAdd(ap + 2, norm * v.z);
    atomicAdd(ap + 3, norm * v.w);
}

// out = relu(agg + dis[i]^2 * hx + b)
__global__ void finalize_kernel(const float* __restrict__ agg,
                                const float* __restrict__ hx,
                                const float* __restrict__ dis,
                                const float* __restrict__ b,
                                float* __restrict__ out, int N) {
    long t = (long)blockIdx.x * blockDim.x + threadIdx.x;
    int i = (int)(t >> 5);
    int lane = (int)(t & 31);
    if (i >= N) return;
    float d2 = dis[i] * dis[i];
    float4 a  = ((const float4*)(agg + (long)i * Hh))[lane];
    float4 x  = ((const float4*)(hx  + (long)i * Hh))[lane];
    float4 bb = ((const float4*)b)[lane];
    float4 r;
    r.x = fmaxf(a.x + d2 * x.x + bb.x, 0.0f);
    r.y = fmaxf(a.y + d2 * x.y + bb.y, 0.0f);
    r.z = fmaxf(a.z + d2 * x.z + bb.z, 0.0f);
    r.w = fmaxf(a.w + d2 * x.w + bb.w, 0.0f);
    ((float4*)(out + (long)i * Hh))[lane] = r;
}

// ---------------------------------------------------------------------------
// pooled[batch[i]] += h[i]; counts[batch[i]] += 1
__global__ void pool_kernel(const float* __restrict__ h,
                            const int* __restrict__ batch,
                            float* __restrict__ pooled,
                            float* __restrict__ counts, int N) {
    long t = (long)blockIdx.x * blockDim.x + threadIdx.x;
    int i = (int)(t >> 5);
    int lane = (int)(t & 31);
    if (i >= N) return;
    int g = batch[i];
    float4 v = ((const float4*)(h + (long)i * Hh))[lane];
    float* pp = pooled + (long)g * Hh + lane * 4;
    atomicAdd(pp + 0, v.x);
    atomicAdd(pp + 1, v.y);
    atomicAdd(pp + 2, v.z);
    atomicAdd(pp + 3, v.w);
    if (lane == 0) atomicAdd(&counts[g], 1.0f);
}

// ---------------------------------------------------------------------------
// out[g] = relu([pooled/cnt, gf] @ Wc1 + bc1) @ Wc2 + bc2   (one thread/graph)
__global__ void classify_kernel(const float* __restrict__ pooled,
                                const float* __restrict__ counts,
                                const float* __restrict__ gf,
                                const float* __restrict__ Wc1,
                                const float* __restrict__ bc1,
                                const float* __restrict__ Wc2,
                                const float* __restrict__ bc2,
                                float* __restrict__ out, int G) {
    int g = blockIdx.x * blockDim.x + threadIdx.x;
    if (g >= G) return;
    float inv = 1.0f / fmaxf(counts[g], 1.0f);
    float o = bc2[0];
    for (int j = 0; j < Hh / 2; ++j) {
        float acc = bc1[j];
        for (int i = 0; i < Hh; ++i)
            acc += pooled[(long)g * Hh + i] * inv * Wc1[(long)i * (Hh / 2) + j];
        for (int i = 0; i < GF; ++i)
            acc += gf[(long)g * GF + i] * Wc1[(long)(Hh + i) * (Hh / 2) + j];
        o += fmaxf(acc, 0.0f) * Wc2[j];
    }
    out[g] = o;
}

// ---------------------------------------------------------------------------
extern "C" void kernel_launch(void* const* d_in, const int* in_sizes, int n_in,
                              void* d_out, int out_size, void* d_ws, size_t ws_size,
                              hipStream_t stream) {
    const float* x   = (const float*)d_in[0];
    const int*   ei  = (const int*)d_in[1];           // [2,E]
    const int*   bat = (const int*)d_in[2];           // [N]
    const float* gf  = (const float*)d_in[3];         // [G,GF]
    const float* W1  = (const float*)d_in[4];
    const float* b1  = (const float*)d_in[5];
    const float* W2  = (const float*)d_in[6];
    const float* b2  = (const float*)d_in[7];
    const float* Wc1 = (const float*)d_in[8];
    const float* bc1 = (const float*)d_in[9];
    const float* Wc2 = (const float*)d_in[10];
    const float* bc2 = (const float*)d_in[11];
    float* out = (float*)d_out;

    const int* src = ei;
    const int* dst = ei + Ee;

    // workspace layout (floats)
    float* ws     = (float*)d_ws;
    float* dis    = ws;                         // N
    float* hx     = dis + Nn;                   // N*H
    float* agg    = hx + (long)Nn * Hh;         // N*H
    float* hbuf   = agg + (long)Nn * Hh;        // N*H
    float* pooled = hbuf + (long)Nn * Hh;       // G*H
    float* counts = pooled + (long)Gg * Hh;     // G

    const int nRowTiles = Nn / 16;              // 3125
    const int gemmBlocks = (nRowTiles * 32 + 127) / 128;
    const long edgeThreads = (long)Ee * 32;
    const long nodeThreads = (long)Nn * 32;

    // degree / normalization
    fill_zero_kernel<<<(Nn + 255) / 256, 256, 0, stream>>>(dis, Nn);
    edge_deg_kernel<<<(Ee + 255) / 256, 256, 0, stream>>>(dst, dis, Ee);
    make_dis_kernel<<<(Nn + 255) / 256, 256, 0, stream>>>(dis, Nn);

    // ---- layer 1 ----
    gemm_f32_wmma_kernel<<<gemmBlocks, 128, 0, stream>>>(x, W1, hx, nRowTiles);
    fill_zero_kernel<<<(int)(((long)Nn * Hh + 255) / 256), 256, 0, stream>>>(agg, (long)Nn * Hh);
    scatter_edges_kernel<<<(int)((edgeThreads + 255) / 256), 256, 0, stream>>>(hx, dis, src, dst, agg, Ee);
    finalize_kernel<<<(int)((nodeThreads + 255) / 256), 256, 0, stream>>>(agg, hx, dis, b1, hbuf, Nn);

    // ---- layer 2 ----
    gemm_f32_wmma_kernel<<<gemmBlocks, 128, 0, stream>>>(hbuf, W2, hx, nRowTiles);
    fill_zero_kernel<<<(int)(((long)Nn * Hh + 255) / 256), 256, 0, stream>>>(agg, (long)Nn * Hh);
    scatter_edges_kernel<<<(int)((edgeThreads + 255) / 256), 256, 0, stream>>>(hx, dis, src, dst, agg, Ee);
    finalize_kernel<<<(int)((nodeThreads + 255) / 256), 256, 0, stream>>>(agg, hx, dis, b2, hbuf, Nn);

    // ---- pool + classifier ----
    fill_zero_kernel<<<(int)(((long)Gg * Hh + Gg + 255) / 256), 256, 0, stream>>>(pooled, (long)Gg * Hh + Gg);
    pool_kernel<<<(int)((nodeThreads + 255) / 256), 256, 0, stream>>>(hbuf, bat, pooled, counts, Nn);
    classify_kernel<<<(Gg + 255) / 256, 256, 0, stream>>>(pooled, counts, gf, Wc1, bc1, Wc2, bc2, out, Gg);
}